// TestModule_31576599561040
// MI455X (gfx1250) — compile-verified
//
#include <hip/hip_runtime.h>

typedef __attribute__((ext_vector_type(2))) float v2f;
typedef __attribute__((ext_vector_type(8))) float v8f;

#define IN_F  5
#define OUT_F 10

// One wave32 processes 16-token tiles with V_WMMA_F32_16X16X4_F32:
//   A (16x4) = x tile (K=0..3), second WMMA adds the K=4 rank-1 term.
//   B (4x16) = dequantized weight^T (padded to N=16), staged in LDS.
//   C init   = bias[n] broadcast (C/D layout: lanes=N, VGPRs=M).
// Fast path (all full tiles): unconditional loads/stores, one loop-invariant
// EXEC mask for the n<10 store columns. Tail tile (at most one, globally)
// handled by a single wave with per-lane guards.
__global__ void __launch_bounds__(256) qlinear_wmma_f32(
    const float* __restrict__ x,
    const float* __restrict__ weight,
    const float* __restrict__ bias,
    float* __restrict__ out,
    int nTok)
{
    __shared__ float wd[16 * 8];   // dequantized weight, row stride 8, rows 10..15 zero
    __shared__ float bs[16];       // bias, entries 10..15 zero

    const int tid = threadIdx.x;

    // ---- one-time 2-bit fake-quant dequant of the 10x5 weight ----
    if (tid < 16) {
        if (tid < OUT_F) {
            float row[IN_F];
            float amax = 0.0f;
#pragma unroll
            for (int k = 0; k < IN_F; ++k) {
                row[k] = weight[tid * IN_F + k];
                amax = fmaxf(amax, fabsf(row[k]));
            }
            // bits=2: qmax = 1, qmin = -2 ; scale = amax/qmax, clamped at 1e-8
            float scale = fmaxf(amax, 1e-8f);
#pragma unroll
            for (int k = 0; k < IN_F; ++k) {
                float q = rintf(row[k] / scale);        // round-half-to-even, like jnp.round
                q = fminf(fmaxf(q, -2.0f), 1.0f);
                wd[tid * 8 + k] = q * scale;
            }
            wd[tid * 8 + 5] = 0.0f;
            wd[tid * 8 + 6] = 0.0f;
            wd[tid * 8 + 7] = 0.0f;
            bs[tid] = bias[tid];
        } else {
#pragma unroll
            for (int k = 0; k < 8; ++k) wd[tid * 8 + k] = 0.0f;
            bs[tid] = 0.0f;
        }
    }
    __syncthreads();

    // ---- per-lane constants (live in VGPRs across the whole tile loop) ----
    const int lane = tid & 31;
    const int n    = lane & 15;     // N index (and A-row M index for loads)
    const int hi   = lane >> 4;     // 0: K=0..1 half, 1: K=2..3 half
    const int k0   = hi << 1;

    v2f b;                          // B matrix: lane holds (K=k0, N=n), (K=k0+1, N=n)
    b.x = wd[n * 8 + k0];
    b.y = wd[n * 8 + k0 + 1];
    v2f b2;                         // second WMMA: K'=0 row = w_deq[:,4]; hi half all-zero
    b2.x = hi ? 0.0f : wd[n * 8 + 4];
    b2.y = 0.0f;
    const float bn = bs[n];

    const int wavesPerBlk = blockDim.x >> 5;
    const int gwave     = blockIdx.x * wavesPerBlk + (tid >> 5);
    const int nwaves    = gridDim.x * wavesPerBlk;
    const int fullTiles = nTok >> 4;                 // tiles entirely in range

    // ================= fast path: full tiles, straight-line =================
    for (int tile = gwave; tile < fullTiles; tile += nwaves) {
        const int t0 = tile << 4;
        const float* p = x + (t0 + n) * IN_F;

        v2f a;  a.x = p[k0]; a.y = p[k0 + 1];
        // hi lanes multiply against b2==0, so the unconditional load is harmless
        v2f a2; a2.x = p[4]; a2.y = 0.0f;

        // branchless prefetch of this wave's next tile
        const int tn  = tile + nwaves;
        const int tpf = (tn < fullTiles) ? tn : tile;
        __builtin_prefetch(x + ((tpf << 4) + n) * IN_F, 0, 3);

        v8f c = { bn, bn, bn, bn, bn, bn, bn, bn };   // bias folded into accumulator
        c = __builtin_amdgcn_wmma_f32_16x16x4_f32(false, a,  false, b,  (short)0, c, false, false);
        c = __builtin_amdgcn_wmma_f32_16x16x4_f32(false, a2, false, b2, (short)0, c, false, false);

        // D layout: VGPR r -> M = r + 8*hi, lane -> N = n.
        if (n < OUT_F) {
            float* q = out + (t0 + (hi << 3)) * OUT_F + n;
#pragma unroll
            for (int r = 0; r < 8; ++r)
                q[r * OUT_F] = c[r];                  // immediate offsets 0..280
        }
    }

    // ================= tail: at most one partial tile, one wave =================
    if ((nTok & 15) && (gwave == (fullTiles % (nwaves ? nwaves : 1)))) {
        const int t0  = fullTiles << 4;
        const int tok = t0 + n;

        v2f a;  a.x = 0.0f; a.y = 0.0f;
        v2f a2; a2.x = 0.0f; a2.y = 0.0f;
        if (tok < nTok) {
            const float* p = x + tok * IN_F;
            a.x  = p[k0];
            a.y  = p[k0 + 1];
            a2.x = p[4];
        }

        v8f c = { bn, bn, bn, bn, bn, bn, bn, bn };
        c = __builtin_amdgcn_wmma_f32_16x16x4_f32(false, a,  false, b,  (short)0, c, false, false);
        c = __builtin_amdgcn_wmma_f32_16x16x4_f32(false, a2, false, b2, (short)0, c, false, false);

        if (n < OUT_F) {
            const int m0 = t0 + (hi << 3);
#pragma unroll
            for (int r = 0; r < 8; ++r) {
                const int m = m0 + r;
                if (m < nTok) out[m * OUT_F + n] = c[r];
            }
        }
    }
}

extern "C" void kernel_launch(void* const* d_in, const int* in_sizes, int n_in,
                              void* d_out, int out_size, void* d_ws, size_t ws_size,
                              hipStream_t stream) {
    const float* x      = (const float*)d_in[0];
    const float* weight = (const float*)d_in[1];
    const float* bias   = (const float*)d_in[2];
    float* out          = (float*)d_out;

    const int nTok   = in_sizes[0] / IN_F;          // 4,000,000
    const int ntiles = (nTok + 15) / 16;            // 250,000 wave-tiles
    int blocks = (ntiles + 7) / 8;                  // 8 waves per 256-thread block
    if (blocks > 2048) blocks = 2048;               // ~16k waves, ~15 tiles each
    if (blocks < 1) blocks = 1;

    qlinear_wmma_f32<<<blocks, 256, 0, stream>>>(x, weight, bias, out, nTok);
}